// GroupCPU_28063316312779
// MI455X (gfx1250) — compile-verified
//
#include <hip/hip_runtime.h>

#define NBATCH 16
#define NPTS   16384
#define NGROUP 1024
#define GSIZE  32

typedef __attribute__((ext_vector_type(2))) float v2f;
typedef __attribute__((ext_vector_type(4))) float v4f;
typedef __attribute__((ext_vector_type(8))) float v8f;

// ---------------------------------------------------------------------------
// Kernel 1: per-point squared norms -> workspace (B*N floats)
// ---------------------------------------------------------------------------
__global__ void pnorm_kernel(const float* __restrict__ xyz,
                             float* __restrict__ pn, int total) {
  int i = blockIdx.x * blockDim.x + threadIdx.x;
  if (i < total) {
    float x = xyz[3 * i + 0];
    float y = xyz[3 * i + 1];
    float z = xyz[3 * i + 2];
    pn[i] = x * x + y * y + z * z;
  }
}

// ---------------------------------------------------------------------------
// Kernel 2: farthest point sampling. One block (1024 threads) per batch.
// The whole batch's point cloud (192 KB) is staged into CDNA5's 320 KB LDS
// once; each of the 1024 sequential steps reads points via conflict-free
// ds_loads. dist[16] is per-thread register state; argmax via LDS tree
// reduction with lowest-index tie-break. Only 16 workgroups exist
// (1 per WGP), so the big LDS footprint is free.
// ---------------------------------------------------------------------------
__global__ __launch_bounds__(1024) void fps_kernel(
    const float* __restrict__ xyz, float* __restrict__ centers) {
  __shared__ float spts[NPTS * 3];   // 192 KB point cache
  __shared__ float rv[1024];
  __shared__ int   ri[1024];
  const int b = blockIdx.x;
  const int t = threadIdx.x;
  const float* p = xyz + (size_t)b * NPTS * 3;

  for (int i = t; i < NPTS * 3; i += 1024) spts[i] = p[i];
  __syncthreads();

  float dist[16];
#pragma unroll
  for (int i = 0; i < 16; ++i) dist[i] = 1.0e10f;

  int far = 0;
  for (int k = 0; k < NGROUP; ++k) {
    float cx = spts[3 * far + 0];
    float cy = spts[3 * far + 1];
    float cz = spts[3 * far + 2];
    if (t == 0) {
      size_t co = ((size_t)b * NGROUP + k) * 3;
      centers[co + 0] = cx;
      centers[co + 1] = cy;
      centers[co + 2] = cz;
    }
    float bestv = -1.0f;
    int   besti = 0;
#pragma unroll
    for (int i = 0; i < 16; ++i) {
      int n = i * 1024 + t;
      float dx = spts[3 * n + 0] - cx;
      float dy = spts[3 * n + 1] - cy;
      float dz = spts[3 * n + 2] - cz;
      float d  = dx * dx + dy * dy + dz * dz;
      dist[i]  = fminf(dist[i], d);
      if (dist[i] > bestv) { bestv = dist[i]; besti = n; }
    }
    rv[t] = bestv;
    ri[t] = besti;
    __syncthreads();
    for (int s = 512; s > 0; s >>= 1) {
      if (t < s) {
        float ov = rv[t + s];
        int   oi = ri[t + s];
        if (ov > rv[t] || (ov == rv[t] && oi < ri[t])) { rv[t] = ov; ri[t] = oi; }
      }
      __syncthreads();
    }
    far = ri[0];
    __syncthreads();
  }
}

// ---------------------------------------------------------------------------
// Kernel 3: kNN (top-32 by squared distance) + gather, WMMA-based.
// One wave (32 lanes) per 32 centers; 512 blocks total.
// Selection key = |p|^2 - 2 c.p   (|c|^2 is constant per row -> dropped).
// c.p tiles come from V_WMMA_F32_16X16X4_F32 (K=3 padded to 4 with zeros).
//
// A (16x4 f32, 2 VGPRs): lanes 0-15 hold K=0,1 ; lanes 16-31 hold K=2,3.
// B (4x16 f32, 2 VGPRs): same K split, lane%16 = column N.
// D (16x16 f32, 8 VGPRs): lane%16 = N, row M = vgpr + 8*(lane/16).
//
// Top-32 list lives in registers. Candidate loops are forced ROLLED
// (#pragma unroll 1) so the unrolled cndmask shift-insert body appears
// exactly once in the code (small loop body -> I-cache resident); only the
// j-indexed shift loops unroll (needed for constant register indexing).
// ---------------------------------------------------------------------------
__global__ __launch_bounds__(32) void knn_gather_kernel(
    const float* __restrict__ xyz, const float* __restrict__ pn,
    const float* __restrict__ centers, float* __restrict__ neigh) {
  __shared__ __align__(16) float keybuf[32][16];  // 32 centers x 16 keys

  const int b     = blockIdx.x >> 5;          // 32 center-tiles per batch
  const int gBase = (blockIdx.x & 31) * 32;
  const int l     = threadIdx.x;              // 0..31
  const int half  = l >> 4;                   // 0: K=0,1   1: K=2,3
  const int mcol  = l & 15;

  const float* p   = xyz + (size_t)b * NPTS * 3;
  const float* pnb = pn + (size_t)b * NPTS;
  const float* cb  = centers + ((size_t)b * NGROUP + gBase) * 3;

  // register-resident sorted top-32 (ascending key)
  float lk[32];
  int   li[32];
#pragma unroll
  for (int j = 0; j < 32; ++j) {
    lk[j] = 3.4e38f;
    li[j] = 0;
  }
  float thr = 3.4e38f;  // == lk[31]

  // A matrices for two 16-center tiles (loop-invariant)
  v2f a0, a1;
  if (half == 0) {
    a0[0] = cb[3 * mcol + 0];        a0[1] = cb[3 * mcol + 1];
    a1[0] = cb[3 * (16 + mcol) + 0]; a1[1] = cb[3 * (16 + mcol) + 1];
  } else {
    a0[0] = cb[3 * mcol + 2];        a0[1] = 0.0f;
    a1[0] = cb[3 * (16 + mcol) + 2]; a1[1] = 0.0f;
  }
  // this lane's own center (for the final gather/subtract)
  const float mycx = cb[3 * l + 0];
  const float mycy = cb[3 * l + 1];
  const float mycz = cb[3 * l + 2];

#pragma unroll 1
  for (int pt = 0; pt < NPTS / 16; ++pt) {
    const int pbase = pt * 16;

    // prefetch next tile's point data (global_prefetch_b8)
    if (pt + 1 < NPTS / 16) {
      __builtin_prefetch(&p[3 * (pbase + 16 + mcol)], 0, 3);
    }

    v2f bv;
    if (half == 0) {
      bv[0] = p[3 * (pbase + mcol) + 0];
      bv[1] = p[3 * (pbase + mcol) + 1];
    } else {
      bv[0] = p[3 * (pbase + mcol) + 2];
      bv[1] = 0.0f;
    }
    const float pnv = pnb[pbase + mcol];

    v8f zero = {0.f, 0.f, 0.f, 0.f, 0.f, 0.f, 0.f, 0.f};
    v8f d0 = __builtin_amdgcn_wmma_f32_16x16x4_f32(
        false, a0, false, bv, (short)0, zero, false, false);
    v8f d1 = __builtin_amdgcn_wmma_f32_16x16x4_f32(
        false, a1, false, bv, (short)0, zero, false, false);

    __syncthreads();  // keybuf readers from previous tile are done
#pragma unroll
    for (int k = 0; k < 8; ++k) {
      const int row = k + 8 * half;
      keybuf[row][mcol]      = pnv - 2.0f * d0[k];
      keybuf[16 + row][mcol] = pnv - 2.0f * d1[k];
    }
    __syncthreads();

    // lane l owns center gBase+l: 16 candidates, fetched as 4x ds_load_b128.
    // Candidate loops stay ROLLED so the insert body exists once.
#pragma unroll 1
    for (int q = 0; q < 4; ++q) {
      const v4f v = ((const v4f*)(&keybuf[l][0]))[q];
#pragma unroll 1
      for (int c2 = 0; c2 < 4; ++c2) {
        const float key = (c2 == 0) ? v[0]
                        : (c2 == 1) ? v[1]
                        : (c2 == 2) ? v[2]
                                    : v[3];
        if (key < thr) {  // common case: compare + skip
          const int idx = pbase + 4 * q + c2;
          bool g[32];
#pragma unroll
          for (int j = 0; j < 32; ++j) g[j] = lk[j] > key;  // vs old keys
#pragma unroll
          for (int j = 31; j > 0; --j) {
            lk[j] = g[j - 1] ? lk[j - 1] : lk[j];
            li[j] = g[j - 1] ? li[j - 1] : li[j];
          }
#pragma unroll
          for (int j = 0; j < 32; ++j) {
            const bool ins = g[j] && (j == 0 || !g[j - 1]);
            lk[j] = ins ? key : lk[j];
            li[j] = ins ? idx : li[j];
          }
          thr = lk[31];
        }
      }
    }
  }

  // gather: neighborhood[b][g][j][:] = xyz[idx] - center
  const int g = gBase + l;
  float* outg = neigh + ((size_t)b * NGROUP + g) * GSIZE * 3;
#pragma unroll 1
  for (int j = 0; j < GSIZE; ++j) {
    const int idx = li[j];
    outg[3 * j + 0] = p[3 * idx + 0] - mycx;
    outg[3 * j + 1] = p[3 * idx + 1] - mycy;
    outg[3 * j + 2] = p[3 * idx + 2] - mycz;
  }
}

// ---------------------------------------------------------------------------
extern "C" void kernel_launch(void* const* d_in, const int* in_sizes, int n_in,
                              void* d_out, int out_size, void* d_ws, size_t ws_size,
                              hipStream_t stream) {
  (void)in_sizes; (void)n_in; (void)out_size; (void)ws_size;
  const float* xyz = (const float*)d_in[0];
  float* out       = (float*)d_out;
  float* neigh     = out;                                       // B*NG*GS*3
  float* centers   = out + (size_t)NBATCH * NGROUP * GSIZE * 3; // B*NG*3
  float* pn        = (float*)d_ws;                              // B*N floats

  const int total = NBATCH * NPTS;
  pnorm_kernel<<<(total + 255) / 256, 256, 0, stream>>>(xyz, pn, total);
  fps_kernel<<<NBATCH, 1024, 0, stream>>>(xyz, centers);
  knn_gather_kernel<<<NBATCH * (NGROUP / 32), 32, 0, stream>>>(xyz, pn, centers, neigh);
}